// CRF_64604898066560
// MI455X (gfx1250) — compile-verified
//
#include <hip/hip_runtime.h>
#include <hip/hip_bf16.h>

typedef __attribute__((ext_vector_type(2))) float v2f;
typedef __attribute__((ext_vector_type(8))) float v8f;

#define Bc 2
#define Hc 80
#define Wc 80
#define Nc (Hc * Wc)          // 6400
#define JT (Nc / 16)          // 400 j-tiles
#define ITERS 5
#define CLIPV 1e-5f

#define INV_2SG2 (1.0f / 18.0f)    // 1/(2*3^2)   gaussian spatial
#define INV_2SB2 (1.0f / 5000.0f)  // 1/(2*50^2)  bilateral spatial
#define INV_2SR2 (1.0f / 50.0f)    // 1/(2*5^2)   bilateral color

// ---------------------------------------------------------------- init
__global__ __launch_bounds__(256) void crf_init(
    const float* __restrict__ img, const float* __restrict__ pred,
    float* __restrict__ posF, float* __restrict__ p2,
    float* __restrict__ colF, float* __restrict__ c2,
    float* __restrict__ U, float* __restrict__ Q)
{
    int t = blockIdx.x * 256 + threadIdx.x;
    if (t >= Bc * Nc) return;
    int b = t / Nc, i = t % Nc;
    int x = i % Wc, y = i / Wc;
    if (b == 0) {
        float fx = (float)x, fy = (float)y;
        posF[4*i+0] = fx; posF[4*i+1] = fy; posF[4*i+2] = 0.f; posF[4*i+3] = 0.f;
        p2[i] = fx*fx + fy*fy;
    }
    float r  = img[((b*3+0)*Hc + y)*Wc + x] * 255.f;
    float g  = img[((b*3+1)*Hc + y)*Wc + x] * 255.f;
    float bl = img[((b*3+2)*Hc + y)*Wc + x] * 255.f;
    colF[4*t+0] = r; colF[4*t+1] = g; colF[4*t+2] = bl; colF[4*t+3] = 0.f;
    c2[t] = r*r + g*g + bl*bl;
    float p  = pred[t];
    float p0 = fminf(fmaxf(p,       CLIPV), 1.f);
    float p1 = fminf(fmaxf(1.f - p, CLIPV), 1.f);
    U[2*t+0] = -__logf(p0);
    U[2*t+1] = -__logf(p1);
    float inv = 1.f / (p0 + p1);       // softmax(-U) == probs / sum(probs)
    Q[2*t+0] = p0 * inv; Q[2*t+1] = p1 * inv;
}

// ------------------------------------------------- fused WMMA tile pass
// Both batches handled in one sweep: posGram / sp2 / expg are shared.
// outg[b,i,l] = sum_j exp(-sp2/18) * srcg[b,j,l]
// outb[b,i,l] = sum_j exp(-sp2/5000 - cd2_b/50) * srcb[b,j,l]
// sumMode: treat srcg/srcb as all-ones (row sums for normalization).
__global__ __launch_bounds__(256) void crf_tile(
    const float* __restrict__ posF, const float* __restrict__ p2,
    const float* __restrict__ colF, const float* __restrict__ c2,
    const float* __restrict__ srcg, const float* __restrict__ srcb,
    float* __restrict__ outg, float* __restrict__ outb, int sumMode)
{
    const int i0   = blockIdx.x * 16;
    const int lane = threadIdx.x & 31;
    const int wave = threadIdx.x >> 5;
    const int n    = lane & 15;
    const int hi   = lane >> 4;     // selects K pair {0,1} vs {2,3}

    // A operands: 16x4 f32, lane = M row, VGPR pair = K half (ISA 7.12.2)
    v2f posA  = *(const v2f*)(posF + (i0 + n) * 4 + 2*hi);
    v2f colA0 = *(const v2f*)(colF + (0*Nc + i0 + n) * 4 + 2*hi);
    v2f colA1 = *(const v2f*)(colF + (1*Nc + i0 + n) * 4 + 2*hi);

    float p2i[8], c2i0[8], c2i1[8];
#pragma unroll
    for (int v = 0; v < 8; ++v) {
        int m = i0 + v + 8*hi;      // C/D layout: VGPR v -> row v (lo) / v+8 (hi)
        p2i[v]  = p2[m];
        c2i0[v] = c2[0*Nc + m];
        c2i1[v] = c2[1*Nc + m];
    }

    float ag[2][2][8] = {}, ab[2][2][8] = {};   // [batch][label][v]

    for (int jt = wave; jt < JT; jt += 8) {   // uniform trip count: EXEC stays all-1s
        int jn = jt * 16 + n;
        v2f posB  = *(const v2f*)(posF + jn * 4 + 2*hi);          // B = F_j^T
        v2f colB0 = *(const v2f*)(colF + (0*Nc + jn) * 4 + 2*hi);
        v2f colB1 = *(const v2f*)(colF + (1*Nc + jn) * 4 + 2*hi);
        float p2j  = p2[jn];
        float c2j0 = c2[0*Nc + jn];
        float c2j1 = c2[1*Nc + jn];
        float qg[2][2] = {{1.f,1.f},{1.f,1.f}};
        float qb[2][2] = {{1.f,1.f},{1.f,1.f}};
        if (!sumMode) {
#pragma unroll
            for (int b = 0; b < 2; ++b) {
                v2f vg = *(const v2f*)(srcg + (b*Nc + jn) * 2);
                v2f vb = *(const v2f*)(srcb + (b*Nc + jn) * 2);
                qg[b][0] = vg.x; qg[b][1] = vg.y;
                qb[b][0] = vb.x; qb[b][1] = vb.y;
            }
        }
        v8f z = {};
        v8f posG  = __builtin_amdgcn_wmma_f32_16x16x4_f32(
            false, posA, false, posB, (short)0, z, false, false);
        v8f colG0 = __builtin_amdgcn_wmma_f32_16x16x4_f32(
            false, colA0, false, colB0, (short)0, z, false, false);
        v8f colG1 = __builtin_amdgcn_wmma_f32_16x16x4_f32(
            false, colA1, false, colB1, (short)0, z, false, false);
#pragma unroll
        for (int v = 0; v < 8; ++v) {
            float sp2 = fmaxf(p2i[v] + p2j - 2.f * posG[v], 0.f);
            float eg  = __expf(-sp2 * INV_2SG2);               // Kg: batch-shared
            float sb  = sp2 * INV_2SB2;
            float cd0 = fmaxf(c2i0[v] + c2j0 - 2.f * colG0[v], 0.f);
            float cd1 = fmaxf(c2i1[v] + c2j1 - 2.f * colG1[v], 0.f);
            float eb0 = __expf(-(sb + cd0 * INV_2SR2));
            float eb1 = __expf(-(sb + cd1 * INV_2SR2));
            ag[0][0][v] = __builtin_fmaf(eg,  qg[0][0], ag[0][0][v]);
            ag[0][1][v] = __builtin_fmaf(eg,  qg[0][1], ag[0][1][v]);
            ag[1][0][v] = __builtin_fmaf(eg,  qg[1][0], ag[1][0][v]);
            ag[1][1][v] = __builtin_fmaf(eg,  qg[1][1], ag[1][1][v]);
            ab[0][0][v] = __builtin_fmaf(eb0, qb[0][0], ab[0][0][v]);
            ab[0][1][v] = __builtin_fmaf(eb0, qb[0][1], ab[0][1][v]);
            ab[1][0][v] = __builtin_fmaf(eb1, qb[1][0], ab[1][0][v]);
            ab[1][1][v] = __builtin_fmaf(eb1, qb[1][1], ab[1][1][v]);
        }
    }

    // butterfly reduce across the 16 lanes holding the same row (halves independent)
#pragma unroll
    for (int mask = 1; mask <= 8; mask <<= 1) {
#pragma unroll
        for (int b = 0; b < 2; ++b)
#pragma unroll
            for (int l = 0; l < 2; ++l)
#pragma unroll
                for (int v = 0; v < 8; ++v) {
                    ag[b][l][v] += __shfl_xor(ag[b][l][v], mask, 32);
                    ab[b][l][v] += __shfl_xor(ab[b][l][v], mask, 32);
                }
    }

    __shared__ float sAcc[16][8];   // [row][kernel(0=g,1=b)*4 + batch*2 + label]
    if (threadIdx.x < 128) sAcc[threadIdx.x & 15][threadIdx.x >> 4] = 0.f;
    __syncthreads();
    if (n == 0) {
#pragma unroll
        for (int v = 0; v < 8; ++v) {
            int m = v + 8*hi;
#pragma unroll
            for (int b = 0; b < 2; ++b)
#pragma unroll
                for (int l = 0; l < 2; ++l) {
                    atomicAdd(&sAcc[m][0*4 + b*2 + l], ag[b][l][v]);
                    atomicAdd(&sAcc[m][1*4 + b*2 + l], ab[b][l][v]);
                }
        }
    }
    __syncthreads();
    if (threadIdx.x < 128) {
        int m = threadIdx.x & 15;
        int s = threadIdx.x >> 4;     // 0..7
        int ker = s >> 2, b = (s >> 1) & 1, l = s & 1;
        float val = sAcc[m][s];
        int idx = (b*Nc + i0 + m) * 2 + l;
        if (ker == 0) outg[idx] = val;
        else          outb[idx] = val;
    }
}

// -------------------------------------------------------- normalization
__global__ __launch_bounds__(256) void crf_norm(
    const float* __restrict__ outg, const float* __restrict__ outb,
    float* __restrict__ ng, float* __restrict__ nb)
{
    int t = blockIdx.x * 256 + threadIdx.x;
    if (t >= Bc * Nc) return;
    nb[t] = rsqrtf(outb[2*t]);             // per-batch bilateral norm
    if (t < Nc) ng[t] = rsqrtf(outg[2*t]); // Kg shared: take b=0 row sums
}

// -------------------------------------------- per-iteration Q premultiply
__global__ __launch_bounds__(256) void crf_prep(
    const float* __restrict__ Q, const float* __restrict__ ng,
    const float* __restrict__ nb, float* __restrict__ qg, float* __restrict__ qb)
{
    int t = blockIdx.x * 256 + threadIdx.x;
    if (t >= Bc * Nc) return;
    int i = t % Nc;
    float g = ng[i], bb = nb[t];
    float q0 = Q[2*t+0], q1 = Q[2*t+1];
    qg[2*t+0] = g  * q0; qg[2*t+1] = g  * q1;
    qb[2*t+0] = bb * q0; qb[2*t+1] = bb * q1;
}

// ----------------------------------------------------- mean-field update
__global__ __launch_bounds__(256) void crf_update(
    const float* __restrict__ U, const float* __restrict__ outg,
    const float* __restrict__ outb, const float* __restrict__ ng,
    const float* __restrict__ nb, float* __restrict__ Q)
{
    int t = blockIdx.x * 256 + threadIdx.x;
    if (t >= Bc * Nc) return;
    int i = t % Nc;
    float g = ng[i], bb = nb[t];
    float z0 = -U[2*t+0] + 3.f * (g * outg[2*t+0]) + 10.f * (bb * outb[2*t+0]);
    float z1 = -U[2*t+1] + 3.f * (g * outg[2*t+1]) + 10.f * (bb * outb[2*t+1]);
    float m  = fmaxf(z0, z1);
    float e0 = __expf(z0 - m), e1 = __expf(z1 - m);
    float inv = 1.f / (e0 + e1);
    Q[2*t+0] = e0 * inv; Q[2*t+1] = e1 * inv;
}

// --------------------------------------------------------------- output
__global__ __launch_bounds__(256) void crf_out(
    const float* __restrict__ Q, float* __restrict__ out)
{
    int t = blockIdx.x * 256 + threadIdx.x;
    if (t >= Bc * Nc) return;
    out[t] = Q[2*t];   // P(label 0), layout B x 1 x H x W
}

// ---------------------------------------------------------------- host
extern "C" void kernel_launch(void* const* d_in, const int* in_sizes, int n_in,
                              void* d_out, int out_size, void* d_ws, size_t ws_size,
                              hipStream_t stream)
{
    const float* img  = (const float*)d_in[0];
    const float* pred = (const float*)d_in[1];
    float* out = (float*)d_out;

    float* ws   = (float*)d_ws;
    float* posF = ws; ws += Nc * 4;
    float* p2   = ws; ws += Nc;
    float* colF = ws; ws += Bc * Nc * 4;
    float* c2   = ws; ws += Bc * Nc;
    float* U    = ws; ws += Bc * Nc * 2;
    float* Q    = ws; ws += Bc * Nc * 2;
    float* qg   = ws; ws += Bc * Nc * 2;
    float* qb   = ws; ws += Bc * Nc * 2;
    float* ng   = ws; ws += Nc;
    float* nb   = ws; ws += Bc * Nc;
    float* outg = ws; ws += Bc * Nc * 2;
    float* outb = ws; ws += Bc * Nc * 2;

    const int nPix = Bc * Nc;
    const int blk  = (nPix + 255) / 256;
    const dim3 tgrid(Nc / 16);

    crf_init<<<blk, 256, 0, stream>>>(img, pred, posF, p2, colF, c2, U, Q);
    // row sums for symmetric kernel normalization (q == 1)
    crf_tile<<<tgrid, 256, 0, stream>>>(posF, p2, colF, c2, Q, Q, outg, outb, 1);
    crf_norm<<<blk, 256, 0, stream>>>(outg, outb, ng, nb);

    for (int it = 0; it < ITERS; ++it) {
        crf_prep<<<blk, 256, 0, stream>>>(Q, ng, nb, qg, qb);
        crf_tile<<<tgrid, 256, 0, stream>>>(posF, p2, colF, c2, qg, qb, outg, outb, 0);
        crf_update<<<blk, 256, 0, stream>>>(U, outg, outb, ng, nb, Q);
    }
    crf_out<<<blk, 256, 0, stream>>>(Q, out);
}